// PSHScatterLayer_12627203851177
// MI455X (gfx1250) — compile-verified
//
#include <hip/hip_runtime.h>
#include <stdint.h>

#define BUCKET_SIZE 512

// PSH primes
#define P1u 73856093u
#define P2u 19349663u
#define P3u 83492791u
#define P4u 2654435761u

#define CHUNK   1024          // points per TDM staging buffer
#define CHUNK3  (CHUNK * 3)   // floats per staging buffer (12 KB)

typedef uint32_t u32x4 __attribute__((ext_vector_type(4)));
typedef uint32_t u32x8 __attribute__((ext_vector_type(8)));

// LDS byte offset (wave-relative, same view DS ops / TDM use)
static __device__ __forceinline__ uint32_t lds_addr_of(const void* p) {
    return (uint32_t)(uintptr_t)(__attribute__((address_space(3))) const void*)p;
}

// Issue a 1-D TDM load: n_tile 4-byte elements from global gptr to the LDS byte address lds_byte_addr.
// n_tensor = valid elements from gptr to end of tensor (OOB reads return 0).
static __device__ __forceinline__ void tdm_load_1d(uint32_t lds_byte_addr,
                                                   const void* gptr,
                                                   uint32_t n_tile,
                                                   uint32_t n_tensor) {
    const uint64_t ga = (uint64_t)(uintptr_t)gptr;
    u32x4 g0;
    g0.x = 1u;                                              // count=1 (valid D#)
    g0.y = lds_byte_addr;                                   // lds_addr
    g0.z = (uint32_t)ga;                                    // global_addr[31:0]
    g0.w = ((uint32_t)(ga >> 32) & 0x1FFFFFFu) | (2u << 30);// global_addr[56:32] | type=2
    u32x8 g1;
    g1.s0 = 2u << 16;                                       // data_size = 4 bytes
    g1.s1 = (n_tensor & 0xFFFFu) << 16;                     // tensor_dim0[15:0]
    g1.s2 = (n_tensor >> 16) | (1u << 16);                  // tensor_dim0[31:16] | tensor_dim1=1
    g1.s3 = n_tile << 16;                                   // tile_dim0 (16-bit)
    g1.s4 = 1u;                                             // tile_dim1=1, tile_dim2=0
    g1.s5 = n_tensor;                                       // tensor_dim0_stride[31:0]
    g1.s6 = 0u;                                             // stride hi / dim1_stride lo
    g1.s7 = 0u;
    asm volatile("tensor_load_to_lds %0, %1" :: "s"(g0), "s"(g1) : "memory");
}

// ---------------------------------------------------------------------------
// K0: zero the scattered-coordinate region (pad_to*3 floats, multiple of 4)
// ---------------------------------------------------------------------------
__global__ void psh_zero(float4* __restrict__ out, int n4) {
    int i = blockIdx.x * blockDim.x + threadIdx.x;
    if (i < n4) out[i] = make_float4(0.f, 0.f, 0.f, 0.f);
}

// ---------------------------------------------------------------------------
// K1: per-segment buckets + stable within-segment ranks + histogram.
// One wave per segment, TDM double-buffered coordinate staging in LDS.
// ---------------------------------------------------------------------------
__global__ void __launch_bounds__(32)
psh_bucket_hist(const float* __restrict__ coords,
                const int*   __restrict__ seps, int B,
                const int*   __restrict__ hash_op_p,
                float*       __restrict__ bucket_out,   // [N] as float
                uint32_t*    __restrict__ packed_ws,    // [N] bucket | rank<<16
                uint32_t*    __restrict__ seg_hist,     // [nsegs][nb]
                int N, int nb, int seg_pts)
{
    extern __shared__ uint32_t lds[];
    uint32_t* hist = lds;                                   // nb words
    const uint32_t stg_words = (uint32_t)((nb + 3) & ~3);   // 16B-align staging
    float* stg0 = (float*)(lds + stg_words);
    float* stg1 = stg0 + CHUNK3;

    const int lane = threadIdx.x;
    const int seg  = blockIdx.x;

    for (int b = lane; b < nb; b += 32) hist[b] = 0;        // same-wave LDS: in order

    const uint32_t hop = (uint32_t)hash_op_p[0];
    const int base_i = seg * seg_pts;
    int remain = N - base_i;
    if (remain <= 0) return;
    if (remain > seg_pts) remain = seg_pts;
    const int chunks = (remain + CHUNK - 1) / CHUNK;
    const size_t total_elems = (size_t)N * 3;

    const uint32_t stg_addr0 = lds_addr_of(stg0);
    const uint32_t stg_addr1 = lds_addr_of(stg1);

    // preload chunk 0
    {
        size_t eb = (size_t)base_i * 3;
        tdm_load_1d(stg_addr0, coords + eb, CHUNK3, (uint32_t)(total_elems - eb));
    }

    for (int c = 0; c < chunks; ++c) {
        const int chunk_base = base_i + c * CHUNK;
        // prefetch next chunk into the other buffer (TDM, async)
        if (c + 1 < chunks) {
            size_t eb = (size_t)(chunk_base + CHUNK) * 3;
            tdm_load_1d(((c + 1) & 1) ? stg_addr1 : stg_addr0,
                        coords + eb, CHUNK3, (uint32_t)(total_elems - eb));
            __builtin_amdgcn_s_wait_tensorcnt(1);           // chunk c complete
        } else {
            __builtin_amdgcn_s_wait_tensorcnt(0);           // last chunk complete
        }
        asm volatile("" ::: "memory");

        const float* stg = (c & 1) ? stg1 : stg0;
        const int pts = (remain - c * CHUNK < CHUNK) ? (remain - c * CHUNK) : CHUNK;
        const int iters = (pts + 31) >> 5;

        for (int it = 0; it < iters; ++it) {
            const int p   = (it << 5) + lane;               // point within chunk
            const int idx = chunk_base + p;
            const bool active = (p < pts);

            if (active) {
                float x = stg[p * 3 + 0];
                float y = stg[p * 3 + 1];
                float z = stg[p * 3 + 2];
                uint32_t ux = (uint32_t)(int)floorf(x);
                uint32_t uy = (uint32_t)(int)floorf(y);
                uint32_t uz = (uint32_t)(int)floorf(z);
                uint32_t batch = 0;
                for (int k = 0; k < B; ++k)                 // seps uniform -> s_loads
                    batch += (uint32_t)(seps[k] <= idx);
                uint32_t h = (ux * P1u) ^ (uy * P2u) ^ (uz * P3u) ^ (batch * P4u);
                h += hop;
                uint32_t bkt = h % (uint32_t)nb;            // exact u32 urem
                bucket_out[idx] = (float)bkt;

                // wave32 match of same-bucket lanes within the active set
                unsigned mask = (unsigned)__ballot(1);
                for (int bit = 0; bit < 16; ++bit) {        // nb < 2^16
                    unsigned bb = (unsigned)__ballot((int)((bkt >> bit) & 1u));
                    mask &= ((bkt >> bit) & 1u) ? bb : ~bb;
                }
                const unsigned lt    = (1u << lane) - 1u;
                const unsigned prev  = __popc(mask & lt);   // earlier same-bucket lanes
                const unsigned total = __popc(mask);

                uint32_t basec = hist[bkt];                 // wave-wide ds_load first
                uint32_t local = basec + prev;              // stable rank in segment
                if (prev == 0) hist[bkt] = basec + total;   // leaders: distinct buckets
                if (local > 0xFFFFu) local = 0xFFFFu;       // >=512 drops anyway
                packed_ws[idx] = bkt | (local << 16);
            }
            __builtin_amdgcn_wave_barrier();                // keep LDS order across iters
        }
    }

    const size_t row = (size_t)seg * (size_t)nb;
    for (int b = lane; b < nb; b += 32) seg_hist[row + b] = hist[b];
}

// ---------------------------------------------------------------------------
// K2: per-bucket exclusive scan over segments (coalesced across buckets),
// plus total counts output (as float, exact at these magnitudes).
// ---------------------------------------------------------------------------
__global__ void psh_scan(uint32_t* __restrict__ seg_hist,
                         float*    __restrict__ counts_out,
                         int nb, int nsegs)
{
    int b = blockIdx.x * blockDim.x + threadIdx.x;
    if (b >= nb) return;
    uint32_t run = 0;
    size_t idx = (size_t)b;
    for (int s = 0; s < nsegs; ++s, idx += (size_t)nb) {
        if (s + 1 < nsegs)
            __builtin_prefetch(&seg_hist[idx + (size_t)nb], 0, 0); // global_prefetch_b8
        uint32_t v = seg_hist[idx];
        seg_hist[idx] = run;                                // exclusive prefix
        run += v;
    }
    counts_out[b] = (float)run;
}

// ---------------------------------------------------------------------------
// K3: fully parallel scatter using precomputed local ranks + scanned bases.
// ---------------------------------------------------------------------------
__global__ void psh_scatter(const float*    __restrict__ coords,
                            const uint32_t* __restrict__ packed_ws,
                            const uint32_t* __restrict__ seg_hist,
                            float*          __restrict__ out,
                            int N, int nb, int seg_shift)
{
    int i = blockIdx.x * blockDim.x + threadIdx.x;
    if (i >= N) return;
    const uint32_t p     = packed_ws[i];
    const uint32_t bkt   = p & 0xFFFFu;
    const uint32_t local = p >> 16;
    const int seg = i >> seg_shift;
    const uint32_t rank = seg_hist[(size_t)seg * (size_t)nb + bkt] + local;
    if (rank < BUCKET_SIZE) {
        const size_t slot = ((size_t)bkt * BUCKET_SIZE + rank) * 3;
        const size_t ci   = 3 * (size_t)i;
        out[slot + 0] = coords[ci + 0];
        out[slot + 1] = coords[ci + 1];
        out[slot + 2] = coords[ci + 2];
    }
}

// ---------------------------------------------------------------------------
extern "C" void kernel_launch(void* const* d_in, const int* in_sizes, int n_in,
                              void* d_out, int out_size, void* d_ws, size_t ws_size,
                              hipStream_t stream) {
    const float* coords = (const float*)d_in[0];
    const int*   seps   = (const int*)d_in[1];
    const int*   hop    = (const int*)d_in[2];

    const int N = in_sizes[0] / 3;
    const int B = in_sizes[1];
    if (N <= 0) return;

    const long long pad_to = ((long long)N + BUCKET_SIZE - 1) / BUCKET_SIZE * BUCKET_SIZE;
    const int nb = (int)(pad_to / BUCKET_SIZE);

    float* out        = (float*)d_out;
    float* counts_out = out + (size_t)pad_to * 3;
    float* bucket_out = counts_out + nb;

    // pick segment size (power of two, 4K..64K points) so workspace fits
    int seg_shift = 12;
    while (seg_shift < 16) {
        size_t nsg  = ((size_t)N + ((size_t)1 << seg_shift) - 1) >> seg_shift;
        size_t need = (size_t)N * 4 + nsg * (size_t)nb * 4;
        if (need <= ws_size) break;
        ++seg_shift;
    }
    const int seg_pts = 1 << seg_shift;
    const int nsegs   = (N + seg_pts - 1) / seg_pts;

    uint32_t* packed   = (uint32_t*)d_ws;
    uint32_t* seg_hist = packed + N;

    // K0: zero scattered region
    {
        int n4 = (int)(pad_to * 3 / 4);
        int blk = 256, grd = (n4 + blk - 1) / blk;
        psh_zero<<<grd, blk, 0, stream>>>((float4*)out, n4);
    }
    // K1: buckets + local ranks + per-segment histograms (TDM staging)
    {
        size_t lds = ((size_t)((nb + 3) & ~3) + 2 * CHUNK3) * sizeof(uint32_t);
        psh_bucket_hist<<<nsegs, 32, lds, stream>>>(coords, seps, B, hop,
                                                    bucket_out, packed, seg_hist,
                                                    N, nb, seg_pts);
    }
    // K2: scan + counts
    {
        int blk = 256, grd = (nb + blk - 1) / blk;
        psh_scan<<<grd, blk, 0, stream>>>(seg_hist, counts_out, nb, nsegs);
    }
    // K3: scatter
    {
        int blk = 256, grd = (N + blk - 1) / blk;
        psh_scatter<<<grd, blk, 0, stream>>>(coords, packed, seg_hist, out,
                                             N, nb, seg_shift);
    }
}